// Wavelet1DEncoderDecoder_38405597561523
// MI455X (gfx1250) — compile-verified
//
#include <hip/hip_runtime.h>
#include <hip/hip_bf16.h>

// ---------------------------------------------------------------------------
// bior4.4 (CDF 9/7) 3-level DWT, batch 4096 x 20000 f32, pywt 'symmetric'.
//
// Per level, for output index i:
//   cA[i] = sum_{t=0..9} x_sym[2i + t - 8] * KF_LO[t]     (KF = flipped filter)
//   cD[i] = sum_{t=0..9} x_sym[2i + t - 8] * KF_HI[t]
// Cast as WMMA: D[16 rows x 16 outs] = A[16 x 40] * W[40 x 16],
// W banded: W[j,n] = KF[j - 2n] (0 when j-2n outside [0,9]).
// 10 chained V_WMMA_F32_16X16X4_F32 per accumulator, 2 accumulators (lo/hi).
// Memory-bound (~1.1 GB over 3 passes @ 23.3 TB/s); WMMA gives the filtering
// for free, so the kernel is organized around clean, clause-able b64 loads.
// ---------------------------------------------------------------------------

typedef float v2f __attribute__((ext_vector_type(2)));
typedef float v8f __attribute__((ext_vector_type(8)));

// Pre-flipped decomposition filters: KF_XX[t] = DEC_XX[9 - t].
__constant__ float KF_LO[10] = {
    0.03782845550699535f, -0.02384946501937986f, -0.11062440441842342f,
    0.3774028556126536f,   0.8526986790094034f,   0.3774028556126536f,
   -0.11062440441842342f, -0.02384946501937986f,  0.03782845550699535f,
    0.0f};
__constant__ float KF_HI[10] = {
    0.0f,                  0.0f,                 -0.06453888262893856f,
    0.04068941760955867f,  0.4180922732222122f,  -0.7884856164056644f,
    0.4180922732222122f,   0.04068941760955867f, -0.06453888262893856f,
    0.0f};

__device__ __forceinline__ int reflect_idx(int c, int N) {
  // pywt 'symmetric' (half-sample) extension; valid for -N <= c < 2N.
  // Ternaries -> v_cndmask: no EXEC change (WMMA needs EXEC all-ones).
  c = (c < 0)  ? (-1 - c)        : c;
  c = (c >= N) ? (2 * N - 1 - c) : c;
  return c;
}

// One DWT level. Each wave32 computes a 16-row x 16-output tile of both cA
// and cD. Block = 128 threads = 4 waves = 64 output columns.
__global__ __launch_bounds__(128)
void dwt_bior44_level_wmma(const float* __restrict__ in, int N,
                           float* __restrict__ outA, int ldA,
                           float* __restrict__ outD, int ldD,
                           int On) {
  const int lane = threadIdx.x & 31;
  const int wave = threadIdx.x >> 5;
  const int hl   = lane >> 4;      // lane half (0: lanes 0-15, 1: lanes 16-31)
  const int mn   = lane & 15;      // A-row index / B,C,D-column index
  const int rowBase = blockIdx.y << 4;
  const int nBase   = (blockIdx.x << 6) + (wave << 4);

  // --- Precompute banded-filter B fragments (lane-local, tile-invariant). ---
  // B (4x16) layout mirrors A/C/D striping: lane half h, vgpr v holds
  // K = 2h + v; with chunk c the absolute j = 4c + 2h + v.
  v2f bLo[10], bHi[10];
#pragma unroll
  for (int c = 0; c < 10; ++c) {
    const int j0 = 4 * c + 2 * hl;
    const int t0 = j0 - 2 * mn;
    const int t1 = t0 + 1;
    bLo[c].x = (t0 >= 0 && t0 < 10) ? KF_LO[t0] : 0.0f;
    bLo[c].y = (t1 >= 0 && t1 < 10) ? KF_LO[t1] : 0.0f;
    bHi[c].x = (t0 >= 0 && t0 < 10) ? KF_HI[t0] : 0.0f;
    bHi[c].y = (t1 >= 0 && t1 < 10) ? KF_HI[t1] : 0.0f;
  }

  // A fragments: this lane supplies row (rowBase+mn), cols winLo+4c+2h+{0,1}.
  const float* rowPtr = in + (long long)(rowBase + mn) * (long long)N;
  const int winLo = 2 * nBase - 8;   // x_sym window start (x coords)

  // Block-uniform interior test: block spans input cols
  // [128*bx - 8, 128*bx + 127]. blockIdx.x lives in an SGPR, so this is a
  // scalar compare -> s_cbranch; no EXEC masking, clean WMMA paths.
  const int bx = (int)blockIdx.x;
  const bool interiorBlock = (bx >= 1) && (128 * bx + 127 < N);

  v8f accA = {0.f, 0.f, 0.f, 0.f, 0.f, 0.f, 0.f, 0.f};
  v8f accD = {0.f, 0.f, 0.f, 0.f, 0.f, 0.f, 0.f, 0.f};

  v2f a[10];
  if (interiorBlock) {
    // All 10 aligned b64 loads first (col parity is always even, row widths
    // even at every level -> 8B aligned), then 20 chained WMMAs.
#pragma unroll
    for (int c = 0; c < 10; ++c)
      a[c] = *(const v2f*)(rowPtr + winLo + 4 * c + 2 * hl);
  } else {
    // Branch-free symmetric reflection; two b32 loads per chunk.
#pragma unroll
    for (int c = 0; c < 10; ++c) {
      const int col0 = winLo + 4 * c + 2 * hl;
      a[c].x = rowPtr[reflect_idx(col0, N)];
      a[c].y = rowPtr[reflect_idx(col0 + 1, N)];
    }
  }

#pragma unroll
  for (int c = 0; c < 10; ++c) {
    accA = __builtin_amdgcn_wmma_f32_16x16x4_f32(
        false, a[c], false, bLo[c], (short)0, accA, false, false);
    accD = __builtin_amdgcn_wmma_f32_16x16x4_f32(
        false, a[c], false, bHi[c], (short)0, accD, false, false);
  }

  // --- Store C/D tile: VGPR v, lane half h -> row v + 8h, col nBase + mn. ---
  const int n = nBase + mn;
  if (n < On) {
    float* pA = outA + (long long)(rowBase + 8 * hl) * (long long)ldA + n;
    float* pD = outD + (long long)(rowBase + 8 * hl) * (long long)ldD + n;
#pragma unroll
    for (int v = 0; v < 8; ++v) {
      pA[(long long)v * ldA] = accA[v];
      pD[(long long)v * ldD] = accD[v];
    }
  }
}

extern "C" void kernel_launch(void* const* d_in, const int* in_sizes, int n_in,
                              void* d_out, int out_size, void* d_ws,
                              size_t ws_size, hipStream_t stream) {
  (void)n_in; (void)out_size; (void)ws_size;

  const float* x = (const float*)d_in[0];
  float* out = (float*)d_out;

  const int N0  = 20000;
  const int B   = in_sizes[0] / N0;      // 4096
  const int On1 = (N0 + 9) >> 1;         // 10004
  const int On2 = (On1 + 9) >> 1;        // 5006
  const int On3 = (On2 + 9) >> 1;        // 2507
  const int Dt  = 2 * On3 + On2 + On1;   // 20024 (output row width)

  // Workspace: cA1 (B x On1) then cA2 (B x On2); cA3 goes straight to d_out.
  float* cA1 = (float*)d_ws;
  float* cA2 = cA1 + (size_t)B * (size_t)On1;

  const dim3 blk(128, 1, 1);
  const dim3 g1((On1 + 63) / 64, B / 16, 1);
  const dim3 g2((On2 + 63) / 64, B / 16, 1);
  const dim3 g3((On3 + 63) / 64, B / 16, 1);

  // Output layout per row: [cA3 | cD3 | cD2 | cD1]. Same stream => in-order.
  dwt_bior44_level_wmma<<<g1, blk, 0, stream>>>(
      x, N0, cA1, On1, out + (2 * On3 + On2), Dt, On1);   // cD1
  dwt_bior44_level_wmma<<<g2, blk, 0, stream>>>(
      cA1, On1, cA2, On2, out + (2 * On3), Dt, On2);      // cD2
  dwt_bior44_level_wmma<<<g3, blk, 0, stream>>>(
      cA2, On2, out, Dt, out + On3, Dt, On3);             // cA3, cD3
}